// GNN1_79783312490852
// MI455X (gfx1250) — compile-verified
//
#include <hip/hip_runtime.h>

typedef __attribute__((ext_vector_type(2))) float v2f;
typedef __attribute__((ext_vector_type(8))) float v8f;

#define N_DRUG 572
#define DIM    512
#define KNBR   64
#define MAXDEG 32
#define NPAD   576          // 36 * 16, padded row count for WMMA tiling
#define MT     36           // NPAD / 16 row tiles
#define NT     32           // DIM / 16 col tiles

// ---------------------------------------------------------------------------
// Kernel 1: q[NPAD, D] = gather(drug_embed, drug_name) @ Wa   (f32 WMMA)
// One wave per 16x16 output tile. A: 16x4 f32 tile (lane=M, half-wave splits K)
// B: 4x16 f32 tile (lane=N). 128 v_wmma_f32_16x16x4_f32 per tile.
// ---------------------------------------------------------------------------
__global__ __launch_bounds__(32)
void qgemm_kernel(const float* __restrict__ drugW, const float* __restrict__ Wa,
                  const int* __restrict__ drug_name, float* __restrict__ q)
{
    const int lane = threadIdx.x;
    const int half = lane >> 4;          // 0: K pair {0,1}; 1: K pair {2,3}
    const int l16  = lane & 15;
    const int mBase = blockIdx.x * 16;
    const int nBase = blockIdx.y * 16;

    const int mrow = mBase + l16;
    const int gi = drug_name[mrow < N_DRUG ? mrow : (N_DRUG - 1)];
    const float* __restrict__ Arow = drugW + (size_t)gi * DIM;
    const float* __restrict__ Bcol = Wa + nBase + l16;

    v8f c = {};
    for (int kk = 0; kk < DIM; kk += 4) {
        const int ka = kk + 2 * half;
        v2f a, b;
        a.x = Arow[ka];
        a.y = Arow[ka + 1];
        b.x = Bcol[(size_t)ka * DIM];
        b.y = Bcol[(size_t)(ka + 1) * DIM];
        c = __builtin_amdgcn_wmma_f32_16x16x4_f32(false, a, false, b,
                                                  (short)0, c, false, false);
    }
    // C/D layout: VGPR j -> row mBase + 8*half + j, col nBase + l16
    float* outBase = q + (size_t)(mBase + half * 8) * DIM + nBase + l16;
#pragma unroll
    for (int j = 0; j < 8; ++j)
        outBase[(size_t)j * DIM] = c[j];
}

// ---------------------------------------------------------------------------
// Kernel 2: relation attention + weighted entity aggregation, one block/drug.
// ---------------------------------------------------------------------------
__global__ __launch_bounds__(256)
void attn_agg_kernel(const float* __restrict__ q, const float* __restrict__ relaW,
                     const float* __restrict__ entW,
                     const int* __restrict__ adj_rel, const int* __restrict__ adj_tail,
                     float* __restrict__ agg)
{
    __shared__ float qs[DIM];
    __shared__ float attn[KNBR];

    const int n  = blockIdx.x;                    // 0..NPAD-1
    const int ni = n < N_DRUG ? n : (N_DRUG - 1); // clamp for index tables
    const int tid = threadIdx.x;

    qs[tid]       = q[(size_t)n * DIM + tid];
    qs[tid + 256] = q[(size_t)n * DIM + tid + 256];
    __syncthreads();

    const int wave = tid >> 5;   // 8 waves, 8 neighbors each
    const int lane = tid & 31;
    for (int kk = 0; kk < 8; ++kk) {
        const int k = wave * 8 + kk;
        const int rel = adj_rel[ni * KNBR + k];
        const float* __restrict__ rw = relaW + (size_t)rel * DIM;
        float p = 0.f;
        for (int j = lane; j < DIM; j += 32)
            p += qs[j] * rw[j];
        for (int off = 16; off >= 1; off >>= 1)
            p += __shfl_xor(p, off, 32);
        if (lane == 0)
            attn[k] = p * 0.04419417382415922f;   // 1/sqrt(512)
    }
    __syncthreads();

    // softmax over 64 scores on wave 0 (each lane owns 2 scores)
    if (tid < 32) {
        float s0 = attn[tid], s1 = attn[tid + 32];
        float m = fmaxf(s0, s1);
        for (int off = 16; off >= 1; off >>= 1)
            m = fmaxf(m, __shfl_xor(m, off, 32));
        float e0 = __expf(s0 - m), e1 = __expf(s1 - m);
        float s = e0 + e1;
        for (int off = 16; off >= 1; off >>= 1)
            s += __shfl_xor(s, off, 32);
        const float inv = 1.f / s;
        attn[tid]      = e0 * inv;
        attn[tid + 32] = e1 * inv;
    }
    __syncthreads();

    // weighted gather-sum over 64 entity rows; thread owns columns tid, tid+256
    float acc0 = 0.f, acc1 = 0.f;
    for (int k = 0; k < KNBR; ++k) {
        const int t = adj_tail[ni * KNBR + k];
        const float w = attn[k];
        const float* __restrict__ er = entW + (size_t)t * DIM;
        acc0 += w * er[tid];
        acc1 += w * er[tid + 256];
    }
    agg[(size_t)n * DIM + tid]       = acc0;
    agg[(size_t)n * DIM + tid + 256] = acc1;
}

// ---------------------------------------------------------------------------
// Kernel 3: h = relu([agg | drug_e] @ lin_W + b)   (f32 WMMA, K = 1024)
// ---------------------------------------------------------------------------
__global__ __launch_bounds__(32)
void lingemm_kernel(const float* __restrict__ agg, const float* __restrict__ drugW,
                    const int* __restrict__ drug_name,
                    const float* __restrict__ linW, const float* __restrict__ linb,
                    float* __restrict__ h)
{
    const int lane = threadIdx.x;
    const int half = lane >> 4;
    const int l16  = lane & 15;
    const int mBase = blockIdx.x * 16;
    const int nBase = blockIdx.y * 16;

    const int mrow = mBase + l16;
    const int gi = drug_name[mrow < N_DRUG ? mrow : (N_DRUG - 1)];
    const float* __restrict__ Arow0 = agg + (size_t)mrow * DIM;    // k in [0,512)
    const float* __restrict__ Arow1 = drugW + (size_t)gi * DIM;    // k in [512,1024)
    const float* __restrict__ Bcol  = linW + nBase + l16;          // lin_W [1024,512]

    v8f c = {};
    for (int kk = 0; kk < 2 * DIM; kk += 4) {
        const int ka = kk + 2 * half;                 // ka, ka+1 same source half
        const float* __restrict__ Ar = (ka < DIM) ? (Arow0 + ka) : (Arow1 + (ka - DIM));
        v2f a, b;
        a.x = Ar[0];
        a.y = Ar[1];
        b.x = Bcol[(size_t)ka * DIM];
        b.y = Bcol[(size_t)(ka + 1) * DIM];
        c = __builtin_amdgcn_wmma_f32_16x16x4_f32(false, a, false, b,
                                                  (short)0, c, false, false);
    }
    const float bias = linb[nBase + l16];
    float* outBase = h + (size_t)(mBase + half * 8) * DIM + nBase + l16;
#pragma unroll
    for (int j = 0; j < 8; ++j) {
        const float v = c[j] + bias;
        outBase[(size_t)j * DIM] = v > 0.f ? v : 0.f;
    }
}

// ---------------------------------------------------------------------------
// Kernel 4: per-column batch stats over N_DRUG rows -> fused scale/shift
// ---------------------------------------------------------------------------
__global__ __launch_bounds__(256)
void bnstats_kernel(const float* __restrict__ h, const float* __restrict__ gamma,
                    const float* __restrict__ beta,
                    float* __restrict__ scale, float* __restrict__ shift)
{
    const int c = blockIdx.x * 256 + threadIdx.x;   // 2 blocks -> 512 columns
    float s = 0.f, s2 = 0.f;
    for (int nrow = 0; nrow < N_DRUG; ++nrow) {
        const float v = h[(size_t)nrow * DIM + c];
        s += v;
        s2 += v * v;
    }
    const float inv_n = 1.0f / (float)N_DRUG;
    const float mean = s * inv_n;
    const float var  = s2 * inv_n - mean * mean;     // biased, like jnp.var
    const float sc = gamma[c] * rsqrtf(var + 1e-5f);
    scale[c] = sc;
    shift[c] = beta[c] - mean * sc;
}

// ---------------------------------------------------------------------------
// Kernel 5: apply BN -> d_out
// ---------------------------------------------------------------------------
__global__ __launch_bounds__(256)
void bnapply_kernel(const float* __restrict__ h, const float* __restrict__ scale,
                    const float* __restrict__ shift, float* __restrict__ out)
{
    const int idx = blockIdx.x * 256 + threadIdx.x;
    if (idx >= N_DRUG * DIM) return;
    const int c = idx & (DIM - 1);
    out[idx] = h[idx] * scale[c] + shift[c];
}

// ---------------------------------------------------------------------------
// Kernel 6: sequential neighbor-mean smoothing. Column-independent: thread c
// replays the full sequential scan for column c (exact in-place semantics).
// ---------------------------------------------------------------------------
__global__ __launch_bounds__(256)
void smooth_kernel(const int* __restrict__ nbr_idx, const int* __restrict__ nbr_deg,
                   const int* __restrict__ epoch, float* __restrict__ f)
{
    if (*epoch <= 1) return;
    const int c = blockIdx.x * 256 + threadIdx.x;   // 2 blocks -> 512 columns
    for (int i = 0; i < N_DRUG; ++i) {
        const int deg = nbr_deg[i];
        if (deg > 0) {
            const int dd = deg < MAXDEG ? deg : MAXDEG;
            float s = 0.f;
            for (int j = 0; j < dd; ++j) {
                const int pos = nbr_idx[i * MAXDEG + j];
                s += f[(size_t)pos * DIM + c];
            }
            const float avg = s / (float)dd;
            const float fi = f[(size_t)i * DIM + c];
            f[(size_t)i * DIM + c] = (avg + fi) * 0.5f;
        }
    }
}

// ---------------------------------------------------------------------------
extern "C" void kernel_launch(void* const* d_in, const int* in_sizes, int n_in,
                              void* d_out, int out_size, void* d_ws, size_t ws_size,
                              hipStream_t stream)
{
    const float* drugW  = (const float*)d_in[0];
    const float* relaW  = (const float*)d_in[1];
    const float* entW   = (const float*)d_in[2];
    const float* Wa     = (const float*)d_in[3];
    const float* linW   = (const float*)d_in[4];
    const float* linb   = (const float*)d_in[5];
    const float* gamma  = (const float*)d_in[6];
    const float* beta   = (const float*)d_in[7];
    const int* drug_name = (const int*)d_in[8];
    const int* adj_tail  = (const int*)d_in[9];
    const int* adj_rel   = (const int*)d_in[10];
    const int* nbr_idx   = (const int*)d_in[11];
    const int* nbr_deg   = (const int*)d_in[12];
    const int* epoch     = (const int*)d_in[13];

    float* out = (float*)d_out;
    float* ws  = (float*)d_ws;
    float* q     = ws;                         // [NPAD, DIM]
    float* agg   = ws + (size_t)NPAD * DIM;    // [NPAD, DIM]
    float* h     = ws + (size_t)2 * NPAD * DIM;// [NPAD, DIM]
    float* scale = ws + (size_t)3 * NPAD * DIM;// [DIM]
    float* shift = scale + DIM;                // [DIM]

    dim3 gTiles(MT, NT);
    qgemm_kernel<<<gTiles, 32, 0, stream>>>(drugW, Wa, drug_name, q);
    attn_agg_kernel<<<NPAD, 256, 0, stream>>>(q, relaW, entW, adj_rel, adj_tail, agg);
    lingemm_kernel<<<gTiles, 32, 0, stream>>>(agg, drugW, drug_name, linW, linb, h);
    bnstats_kernel<<<2, 256, 0, stream>>>(h, gamma, beta, scale, shift);
    bnapply_kernel<<<(N_DRUG * DIM + 255) / 256, 256, 0, stream>>>(h, scale, shift, out);
    smooth_kernel<<<2, 256, 0, stream>>>(nbr_idx, nbr_deg, epoch, out);
}